// GraphAttentionLayer_18399639896893
// MI455X (gfx1250) — compile-verified
//
#include <hip/hip_runtime.h>
#include <math.h>

#define N_NODES 8192
#define N_EDGES 262144
#define DIN 128
#define DOUT 128
#define ALPHA_LRELU 0.2f

typedef __attribute__((ext_vector_type(2))) float v2f;
typedef __attribute__((ext_vector_type(8))) float v8f;

// ---------------------------------------------------------------------------
// Init accumulators (re-run every call; harness does not re-poison between
// replays, so everything we accumulate into must be zeroed here).
// ---------------------------------------------------------------------------
__global__ void gat_init(float* __restrict__ rowmax, float* __restrict__ rowsum,
                         int* __restrict__ rowcnt, int* __restrict__ cursor,
                         float* __restrict__ T, float* __restrict__ Bsum) {
  int i = blockIdx.x * blockDim.x + threadIdx.x;
  if (i < N_NODES) { rowmax[i] = 0.f; rowsum[i] = 0.f; rowcnt[i] = 0; cursor[i] = 0; }
  if (i < DOUT) T[i] = 0.f;
  if (i == 0) Bsum[0] = 0.f;
}

// ---------------------------------------------------------------------------
// Wh = x @ W  via V_WMMA_F32_16X16X4_F32 (full f32 fidelity).
// One wave per 16x16 output tile; block = 8 waves = the 8 column tiles of one
// 16-row strip. W (64 KB) staged in LDS once per block.
// A layout (16x4 f32): lane l: m=l&15, k_base=(l>>4)*2, vgpr0=K(k_base), vgpr1=K(k_base+1)
// B layout (4x16 f32): lane l: n=l&15, k_base=(l>>4)*2 (mirrors A)
// C/D (16x16 f32): vgpr v: row = v + 8*(l>=16), col = l&15
// ---------------------------------------------------------------------------
__global__ void __launch_bounds__(256) gat_gemm_wmma(const float* __restrict__ x,
                                                     const float* __restrict__ W,
                                                     float* __restrict__ Wh) {
  __shared__ float wlds[DIN * DOUT]; // 64 KB
  const int t = threadIdx.x;
  const float4* W4 = (const float4*)W;
  float4* wl4 = (float4*)wlds;
#pragma unroll
  for (int i = 0; i < 16; ++i) wl4[t + 256 * i] = W4[t + 256 * i];
  __syncthreads();

  const int wave = t >> 5;          // 0..7 -> column tile
  const int lane = t & 31;
  const int tm = blockIdx.x;        // row tile 0..511
  const int m = lane & 15;
  const int kb = (lane >> 4) << 1;  // 0 or 2
  const int half = lane >> 4;
  const float* xrow = x + (tm * 16 + m) * DIN;
  const int bcol = wave * 16 + m;

  v8f acc = {0.f, 0.f, 0.f, 0.f, 0.f, 0.f, 0.f, 0.f};
#pragma unroll 4
  for (int k0 = 0; k0 < DIN; k0 += 4) {
    v2f a = *(const v2f*)(xrow + k0 + kb);
    v2f b;
    b[0] = wlds[(k0 + kb) * DOUT + bcol];
    b[1] = wlds[(k0 + kb + 1) * DOUT + bcol];
    acc = __builtin_amdgcn_wmma_f32_16x16x4_f32(false, a, false, b, (short)0, acc,
                                                false, false);
  }
#pragma unroll
  for (int v = 0; v < 8; ++v)
    Wh[(tm * 16 + v + 8 * half) * DOUT + bcol] = acc[v];
}

// ---------------------------------------------------------------------------
// s[i] = Wh[i]·a[0:128], d[i] = Wh[i]·a[128:256]; one wave per node.
// ---------------------------------------------------------------------------
__global__ void __launch_bounds__(256) gat_sd(const float* __restrict__ Wh,
                                              const float* __restrict__ a,
                                              float* __restrict__ s, float* __restrict__ d) {
  int gid = blockIdx.x * blockDim.x + threadIdx.x;
  int node = gid >> 5;
  int lane = threadIdx.x & 31;
  if (node >= N_NODES) return;
  float4 wv = ((const float4*)(Wh + node * DOUT))[lane];
  float4 vs = ((const float4*)a)[lane];
  float4 vd = ((const float4*)(a + DOUT))[lane];
  float ps = wv.x * vs.x + wv.y * vs.y + wv.z * vs.z + wv.w * vs.w;
  float pd = wv.x * vd.x + wv.y * vd.y + wv.z * vd.z + wv.w * vd.w;
  for (int o = 16; o > 0; o >>= 1) {
    ps += __shfl_down(ps, o, 32);
    pd += __shfl_down(pd, o, 32);
  }
  if (lane == 0) { s[node] = ps; d[node] = pd; }
}

// T = colsum(Wh): block handles 128 rows, coalesced column access, LDS reduce.
__global__ void __launch_bounds__(256) gat_colsum(const float* __restrict__ Wh,
                                                  float* __restrict__ T) {
  __shared__ float part[256];
  int t = threadIdx.x;
  int col = t & 127;
  int grp = t >> 7;
  int r0 = blockIdx.x * 128 + grp * 64;
  float acc = 0.f;
  for (int r = 0; r < 64; ++r) acc += Wh[(r0 + r) * DOUT + col];
  part[t] = acc;
  __syncthreads();
  if (t < 128) atomicAdd(&T[t], part[t] + part[t + 128]);
}

// Edge pass A: logits + leaky relu; per-row max (int-punned atomic, valid for
// positives vs 0-init) and edge counts.
__global__ void gat_edge_a(const int* __restrict__ src, const int* __restrict__ dst,
                           const float* __restrict__ s, const float* __restrict__ d,
                           float* __restrict__ ew, float* __restrict__ rowmax,
                           int* __restrict__ rowcnt) {
  int k = blockIdx.x * blockDim.x + threadIdx.x;
  if (k >= N_EDGES) return;
  int i = src[k], j = dst[k];
  float e = s[i] + d[j];
  e = e > 0.f ? e : ALPHA_LRELU * e;
  ew[k] = e;
  if (e > 0.f) atomicMax((int*)(rowmax + i), __float_as_int(e));
  atomicAdd(rowcnt + i, 1);
}

// Edge pass B: w = exp(e - rowmax); row sums.
__global__ void gat_edge_b(const int* __restrict__ src, const float* __restrict__ rowmax,
                           float* __restrict__ ew, float* __restrict__ rowsum) {
  int k = blockIdx.x * blockDim.x + threadIdx.x;
  if (k >= N_EDGES) return;
  int i = src[k];
  float w = expf(ew[k] - rowmax[i]);
  ew[k] = w;
  atomicAdd(rowsum + i, w);
}

// Exclusive prefix scan of 8192 row counts -> CSR offsets. Single block.
__global__ void __launch_bounds__(1024) gat_scan(const int* __restrict__ cnt,
                                                 int* __restrict__ off) {
  __shared__ int sums[1024];
  int t = threadIdx.x;
  int local[8];
  int s = 0;
#pragma unroll
  for (int i = 0; i < 8; ++i) { local[i] = s; s += cnt[t * 8 + i]; }
  sums[t] = s;
  __syncthreads();
  for (int o = 1; o < 1024; o <<= 1) {
    int v = (t >= o) ? sums[t - o] : 0;
    __syncthreads();
    sums[t] += v;
    __syncthreads();
  }
  int base = (t > 0) ? sums[t - 1] : 0;
#pragma unroll
  for (int i = 0; i < 8; ++i) off[t * 8 + i] = base + local[i];
  if (t == 1023) off[N_NODES] = sums[1023];
}

// Per-row softmax constants: Z = (N-cnt)*exp(-max) + S; base = exp(-max)/Z.
// Also reduce B = sum_i base_i.
__global__ void __launch_bounds__(256) gat_rowfin(const float* __restrict__ rowmax,
                                                  const float* __restrict__ rowsum,
                                                  const int* __restrict__ rowcnt,
                                                  float* __restrict__ basev,
                                                  float* __restrict__ invZ,
                                                  float* __restrict__ Bsum) {
  __shared__ float red[256];
  int i = blockIdx.x * blockDim.x + threadIdx.x;
  float em = expf(-rowmax[i]);
  float Z = (float)(N_NODES - rowcnt[i]) * em + rowsum[i];
  float iz = 1.f / Z;
  invZ[i] = iz;
  float b = em * iz;
  basev[i] = b;
  int t = threadIdx.x;
  red[t] = b;
  __syncthreads();
  for (int o = 128; o > 0; o >>= 1) {
    if (t < o) red[t] += red[t + o];
    __syncthreads();
  }
  if (t == 0) atomicAdd(Bsum, red[0]);
}

// Scatter edges into CSR order keyed by src row.
__global__ void gat_csr(const int* __restrict__ src, const int* __restrict__ dst,
                        const float* __restrict__ ew, const int* __restrict__ off,
                        int* __restrict__ cursor, int* __restrict__ csr_col,
                        float* __restrict__ csr_w) {
  int k = blockIdx.x * blockDim.x + threadIdx.x;
  if (k >= N_EDGES) return;
  int i = src[k];
  int pos = off[i] + atomicAdd(cursor + i, 1);
  csr_col[pos] = dst[k];
  csr_w[pos] = ew[k];
}

__global__ void gat_imp_init(float* __restrict__ out_imp, const float* __restrict__ Bsum) {
  int j = blockIdx.x * blockDim.x + threadIdx.x;
  if (j < N_NODES) out_imp[j] = Bsum[0];
}

__global__ void gat_imp_scatter(const int* __restrict__ src, const int* __restrict__ dst,
                                const float* __restrict__ ew, const float* __restrict__ invZ,
                                const float* __restrict__ basev, float* __restrict__ out_imp) {
  int k = blockIdx.x * blockDim.x + threadIdx.x;
  if (k >= N_EDGES) return;
  int i = src[k];
  atomicAdd(out_imp + dst[k], ew[k] * invZ[i] - basev[i]);
}

// h_prime[i] = base_i*T + sum_edges (att-base_i)*Wh[col]; one wave per row,
// lane covers 4 columns (float4). ELU epilogue. Hot loop: E*512B gathers.
__global__ void __launch_bounds__(256) gat_hprime(const float* __restrict__ Wh,
                                                  const float* __restrict__ T,
                                                  const float* __restrict__ basev,
                                                  const float* __restrict__ invZ,
                                                  const int* __restrict__ off,
                                                  const int* __restrict__ csr_col,
                                                  const float* __restrict__ csr_w,
                                                  float* __restrict__ out_h) {
  int gid = blockIdx.x * blockDim.x + threadIdx.x;
  int row = gid >> 5;
  int lane = threadIdx.x & 31;
  if (row >= N_NODES) return;
  float b = basev[row], iz = invZ[row];
  float4 tv = ((const float4*)T)[lane];
  float ax = b * tv.x, ay = b * tv.y, az = b * tv.z, aw = b * tv.w;
  int p0 = off[row], p1 = off[row + 1];
  const float4* Wh4 = (const float4*)Wh;
  for (int p = p0; p < p1; ++p) {
    int col = csr_col[p];
    float c = csr_w[p] * iz - b;
    float4 wv = Wh4[col * 32 + lane];
    ax += c * wv.x; ay += c * wv.y; az += c * wv.z; aw += c * wv.w;
  }
  float4 o;
  o.x = ax > 0.f ? ax : expm1f(ax);
  o.y = ay > 0.f ? ay : expm1f(ay);
  o.z = az > 0.f ? az : expm1f(az);
  o.w = aw > 0.f ? aw : expm1f(aw);
  ((float4*)out_h)[row * 32 + lane] = o;
}

extern "C" void kernel_launch(void* const* d_in, const int* in_sizes, int n_in,
                              void* d_out, int out_size, void* d_ws, size_t ws_size,
                              hipStream_t stream) {
  const float* x = (const float*)d_in[0];
  const int* edge = (const int*)d_in[1];
  const float* W = (const float*)d_in[2];
  const float* a = (const float*)d_in[3];
  const int* src = edge;
  const int* dst = edge + N_EDGES;
  float* out_h = (float*)d_out;                // N*OUT floats (elu(h_prime))
  float* out_imp = out_h + N_NODES * DOUT;     // N floats (node_importance)

  char* p = (char*)d_ws;
  auto carve = [&](size_t bytes) { char* q = p; p += (bytes + 255) & ~size_t(255); return q; };
  float* Wh     = (float*)carve(sizeof(float) * N_NODES * DOUT);
  float* sbuf   = (float*)carve(sizeof(float) * N_NODES);
  float* dbuf   = (float*)carve(sizeof(float) * N_NODES);
  float* rowmax = (float*)carve(sizeof(float) * N_NODES);
  float* rowsum = (float*)carve(sizeof(float) * N_NODES);
  float* basev  = (float*)carve(sizeof(float) * N_NODES);
  float* invZ   = (float*)carve(sizeof(float) * N_NODES);
  float* T      = (float*)carve(sizeof(float) * DOUT);
  float* Bsum   = (float*)carve(sizeof(float) * 8);
  int*   rowcnt = (int*)carve(sizeof(int) * N_NODES);
  int*   rowoff = (int*)carve(sizeof(int) * (N_NODES + 8));
  int*   cursor = (int*)carve(sizeof(int) * N_NODES);
  float* ew     = (float*)carve(sizeof(float) * N_EDGES);
  int*   csrcol = (int*)carve(sizeof(int) * N_EDGES);
  float* csrw   = (float*)carve(sizeof(float) * N_EDGES);

  gat_init<<<N_NODES / 256, 256, 0, stream>>>(rowmax, rowsum, rowcnt, cursor, T, Bsum);
  gat_gemm_wmma<<<N_NODES / 16, 256, 0, stream>>>(x, W, Wh);
  gat_sd<<<N_NODES / 8, 256, 0, stream>>>(Wh, a, sbuf, dbuf);
  gat_colsum<<<N_NODES / 128, 256, 0, stream>>>(Wh, T);
  gat_edge_a<<<N_EDGES / 256, 256, 0, stream>>>(src, dst, sbuf, dbuf, ew, rowmax, rowcnt);
  gat_edge_b<<<N_EDGES / 256, 256, 0, stream>>>(src, rowmax, ew, rowsum);
  gat_scan<<<1, 1024, 0, stream>>>(rowcnt, rowoff);
  gat_rowfin<<<N_NODES / 256, 256, 0, stream>>>(rowmax, rowsum, rowcnt, basev, invZ, Bsum);
  gat_csr<<<N_EDGES / 256, 256, 0, stream>>>(src, dst, ew, rowoff, cursor, csrcol, csrw);
  gat_imp_init<<<N_NODES / 256, 256, 0, stream>>>(out_imp, Bsum);
  gat_hprime<<<N_NODES / 8, 256, 0, stream>>>(Wh, T, basev, invZ, rowoff, csrcol, csrw, out_h);
  gat_imp_scatter<<<N_EDGES / 256, 256, 0, stream>>>(src, dst, ew, invZ, basev, out_imp);
}